// EPNNModel_39917426049338
// MI455X (gfx1250) — compile-verified
//
#include <hip/hip_runtime.h>

typedef float v2f __attribute__((ext_vector_type(2)));
typedef float v8f __attribute__((ext_vector_type(8)));

#define TOLF 1e-5f
#define CLIPMAXF 1e5f

static constexpr int Bb = 8, Nn = 256;

// ---------------------------------------------------------------------------
// Kernel 1: per-atom layer-1 projections.
//   A[atom][c]  = b1[c] + inp_atom @ W1[0:36, c]     (inp_atom = [x(3), h(32), q(1)])
//   Bp[atom][c] =         inp_atom @ W1[36:72, c]
// 131072 threads: gid>>16 selects A vs Bp, (gid>>5)&2047 = atom, gid&31 = col.
// ---------------------------------------------------------------------------
__global__ __launch_bounds__(256) void k_atom_proj(
    const float* __restrict__ x, const float* __restrict__ h,
    const float* __restrict__ q, const float* __restrict__ W1,
    const float* __restrict__ b1, float* __restrict__ Aws,
    float* __restrict__ Bws) {
  int gid = blockIdx.x * 256 + threadIdx.x;
  int col = gid & 31;
  int a   = (gid >> 5) & 2047;
  int sel = gid >> 16;  // 0 -> A (+b1), 1 -> Bp
  const float* Wc = W1 + (sel ? 36 * 32 : 0) + col;
  float acc = sel ? 0.f : b1[col];
  const float* xv = x + a * 3;
  acc += xv[0] * Wc[0] + xv[1] * Wc[32] + xv[2] * Wc[64];
  const float* hv = h + a * 32;
#pragma unroll
  for (int k = 0; k < 32; ++k) acc += hv[k] * Wc[(3 + k) * 32];
  acc += q[a] * Wc[35 * 32];
  (sel ? Bws : Aws)[a * 32 + col] = acc;
}

// ---------------------------------------------------------------------------
// Kernel 2: edge MLP (layers 2,3 via WMMA f32 16x16x4) + antisymmetrization.
// Grid: (b=8, i-tile=16, j-chunk=4). Block: 256 threads = 8 wave32.
// Each j-tile of 16 is processed in 4 sub-steps of 8 M-tiles (4 i-rows ×
// 2 directions); each wave owns one 16-row M-tile.
// ---------------------------------------------------------------------------
__global__ __launch_bounds__(256) void k_edge_mlp(
    const float* __restrict__ e, const float* __restrict__ mask,
    const float* __restrict__ W1, const float* __restrict__ W2,
    const float* __restrict__ b2, const float* __restrict__ W3,
    const float* __restrict__ Aws, const float* __restrict__ Bws,
    float* __restrict__ partials) {
  __shared__ __align__(16) float sh_h1[8 * 544];   // 8 M-tiles x 16 rows x stride 34
  __shared__ __align__(16) float sh_W2T[32 * 34];  // W2 transposed, stride 34
  __shared__ float sh_AI[16 * 33], sh_BI[16 * 33];
  __shared__ float sh_AJ[16 * 33], sh_BJ[16 * 33];
  __shared__ float sh_W1e[128], sh_W3v[32], sh_b2v[32];
  __shared__ float sh_elec[512];  // [dir][it][jm]

  const int t = threadIdx.x;
  const int bb = blockIdx.x, iT = blockIdx.y, chunk = blockIdx.z;
  const int I0 = iT * 16;
  const int w = t >> 5, lane = t & 31, m = lane & 15, hi = lane >> 4;
  const int dirw = w & 1, itl = w >> 1;

  // ---- one-time loads -----------------------------------------------------
#pragma unroll
  for (int k = 0; k < 4; ++k) {  // W2^T (32x32 -> stride 34)
    int idx = t + k * 256;
    int n = idx >> 5, kk = idx & 31;
    sh_W2T[n * 34 + kk] = W2[kk * 32 + n];
  }
  if (t < 128) sh_W1e[t] = W1[(72 + (t >> 5)) * 32 + (t & 31)];
  if (t < 32) { sh_W3v[t] = W3[t]; sh_b2v[t] = b2[t]; }
#pragma unroll
  for (int k = 0; k < 4; ++k) {  // A / Bp rows for the i-tile
    int idx = t + k * 256;
    int selM = idx >> 9, row = (idx >> 5) & 15, col = idx & 31;
    float v = (selM ? Bws : Aws)[(bb * Nn + I0 + row) * 32 + col];
    (selM ? sh_BI : sh_AI)[row * 33 + col] = v;
  }
  __syncthreads();

  // Per-wave W2 B-operands (reused for every tile): lane holds
  // W2[k][n] with n = 16*tile + m, k = 4s + 2*hi (+1 in .y).
  v2f wbA[8], wbB[8];
#pragma unroll
  for (int s = 0; s < 8; ++s) {
    wbA[s] = *(const v2f*)&sh_W2T[m * 34 + 4 * s + 2 * hi];
    wbB[s] = *(const v2f*)&sh_W2T[(16 + m) * 34 + 4 * s + 2 * hi];
  }
  const float w3a = sh_W3v[m], w3b = sh_W3v[16 + m];
  const float c0i = sh_b2v[m], c1i = sh_b2v[16 + m];

  float qacc = 0.f;

  for (int jt = 0; jt < 4; ++jt) {
    const int J0 = (chunk * 4 + jt) * 16;
#pragma unroll
    for (int k = 0; k < 4; ++k) {  // A / Bp rows for the j-tile
      int idx = t + k * 256;
      int selM = idx >> 9, row = (idx >> 5) & 15, col = idx & 31;
      float v = (selM ? Bws : Aws)[(bb * Nn + J0 + row) * 32 + col];
      (selM ? sh_BJ : sh_AJ)[row * 33 + col] = v;
    }
    __syncthreads();

    for (int sub = 0; sub < 4; ++sub) {
      // ---- phase 1: h1 = relu(A_i + Bp_j + e@W1e), both directions -------
      {
        int itl1 = t >> 6;        // 0..3 local i row
        int m1 = (t >> 2) & 15;   // j offset
        int cg = t & 3;           // column group of 8
        int it = sub * 4 + itl1;
        int i = I0 + it, j = J0 + m1;
        const float4 ev = *(const float4*)&e[(size_t)((bb * Nn + i) * Nn + j) * 4];
        float* h0p = &sh_h1[(itl1 * 2 + 0) * 544 + m1 * 34];
        float* h1p = &sh_h1[(itl1 * 2 + 1) * 544 + m1 * 34];
        const float* ai = &sh_AI[it * 33];
        const float* bi = &sh_BI[it * 33];
        const float* aj = &sh_AJ[m1 * 33];
        const float* bj = &sh_BJ[m1 * 33];
#pragma unroll
        for (int cc = 0; cc < 8; ++cc) {
          int col = cg * 8 + cc;
          float ce = ev.x * sh_W1e[col] + ev.y * sh_W1e[32 + col] +
                     ev.z * sh_W1e[64 + col] + ev.w * sh_W1e[96 + col];
          h0p[col] = fmaxf(ai[col] + bj[col] + ce, 0.f);
          h1p[col] = fmaxf(aj[col] + bi[col] + ce, 0.f);
        }
      }
      __syncthreads();

      // ---- phase 2: layer 2 via V_WMMA_F32_16X16X4_F32 + layer-3 reduce --
      {
        int it = sub * 4 + itl;
        const float* hb = &sh_h1[w * 544 + m * 34];
        v2f a[8];
#pragma unroll
        for (int s = 0; s < 8; ++s) a[s] = *(const v2f*)&hb[4 * s + 2 * hi];
        v8f d0, d1;
#pragma unroll
        for (int v = 0; v < 8; ++v) { d0[v] = c0i; d1[v] = c1i; }
#pragma unroll
        for (int s = 0; s < 8; ++s) {
          d0 = __builtin_amdgcn_wmma_f32_16x16x4_f32(false, a[s], false, wbA[s],
                                                     (short)0, d0, false, false);
          d1 = __builtin_amdgcn_wmma_f32_16x16x4_f32(false, a[s], false, wbB[s],
                                                     (short)0, d1, false, false);
        }
        // layer 3: elec[row] = sum_n relu(h2[row][n]) * W3[n]  (b3 cancels)
        float p[8];
#pragma unroll
        for (int v = 0; v < 8; ++v)
          p[v] = fmaxf(d0[v], 0.f) * w3a + fmaxf(d1[v], 0.f) * w3b;
#pragma unroll
        for (int off = 8; off >= 1; off >>= 1) {
#pragma unroll
          for (int v = 0; v < 8; ++v) p[v] += __shfl_xor(p[v], off, 32);
        }
        if (m == 0) {
#pragma unroll
          for (int v = 0; v < 8; ++v)
            sh_elec[(dirw * 16 + it) * 16 + v + 8 * hi] = p[v];
        }
      }
      __syncthreads();
    }

    // ---- phase 3: antisym accumulation -----------------------------------
    {
      int it3 = t >> 4, jm = t & 15;
      int i = I0 + it3, j = J0 + jm;
      const float4 ev = *(const float4*)&e[(size_t)((bb * Nn + i) * Nn + j) * 4];
      float le = fminf(fmaxf(ev.x, TOLF), CLIPMAXF);
      le = fmaxf(le, fminf(fmaxf(ev.y, TOLF), CLIPMAXF));
      le = fmaxf(le, fminf(fmaxf(ev.z, TOLF), CLIPMAXF));
      le = fmaxf(le, fminf(fmaxf(ev.w, TOLF), CLIPMAXF));
      float nearf = (le != TOLF) ? 1.f : 0.f;
      float mr = mask[(size_t)((bb * Nn + i) * Nn + j)];
      qacc += 0.5f * (sh_elec[it3 * 16 + jm] - sh_elec[256 + it3 * 16 + jm]) *
              mr * nearf;
    }
    __syncthreads();
  }

  // reduce qacc over the 16 jm lanes of each half, write per-chunk partial
#pragma unroll
  for (int off = 8; off >= 1; off >>= 1) qacc += __shfl_xor(qacc, off, 32);
  int it3 = t >> 4, jm = t & 15;
  if (jm == 0) partials[((bb * 16 + iT) * 16 + it3) * 4 + chunk] = qacc;
}

// ---------------------------------------------------------------------------
// Kernel 3: q_out = q + sum of the 4 chunk partials per atom.
// ---------------------------------------------------------------------------
__global__ __launch_bounds__(256) void k_final(
    const float* __restrict__ q, const float* __restrict__ partials,
    float* __restrict__ out) {
  int gid = blockIdx.x * 256 + threadIdx.x;
  if (gid < Bb * Nn) {
    const float* p = partials + gid * 4;
    out[gid] = q[gid] + ((p[0] + p[1]) + (p[2] + p[3]));
  }
}

extern "C" void kernel_launch(void* const* d_in, const int* in_sizes, int n_in,
                              void* d_out, int out_size, void* d_ws,
                              size_t ws_size, hipStream_t stream) {
  (void)in_sizes; (void)n_in; (void)out_size; (void)ws_size;
  const float* h    = (const float*)d_in[0];
  const float* e    = (const float*)d_in[1];
  const float* x    = (const float*)d_in[2];
  const float* q    = (const float*)d_in[3];
  const float* mask = (const float*)d_in[4];
  const float* W1   = (const float*)d_in[5];
  const float* b1   = (const float*)d_in[6];
  const float* W2   = (const float*)d_in[7];
  const float* b2   = (const float*)d_in[8];
  const float* W3   = (const float*)d_in[9];
  // d_in[10] = b3: cancels exactly in the antisymmetrization.

  float* Aws = (float*)d_ws;                 // 2048*32 floats
  float* Bws = Aws + 2048 * 32;              // 2048*32 floats
  float* partials = Bws + 2048 * 32;         // 2048*4 floats
  float* out = (float*)d_out;

  k_atom_proj<<<512, 256, 0, stream>>>(x, h, q, W1, b1, Aws, Bws);
  dim3 grid(Bb, 16, 4);
  k_edge_mlp<<<grid, 256, 0, stream>>>(e, mask, W1, W2, b2, W3, Aws, Bws,
                                       partials);
  k_final<<<Bb, 256, 0, stream>>>(q, partials, out);
}